// GraphSAGE_7550552506693
// MI455X (gfx1250) — compile-verified
//
#include <hip/hip_runtime.h>

#define N_NODES 100000
#define N_EDGES 1600000
#define D 64

typedef __attribute__((ext_vector_type(2))) float v2f;
typedef __attribute__((ext_vector_type(8))) float v8f;

// ---------------------------------------------------------------------------
// Phase 0: zero the accumulation scratch (s[N*64] ++ cnt[N]) in d_ws.
// N*(D+1) = 6.5M floats, divisible by 4 -> float4 stores.
// ---------------------------------------------------------------------------
__global__ void sage_zero_kernel(float4* __restrict__ p, int n4) {
    int i = blockIdx.x * blockDim.x + threadIdx.x;
    if (i < n4) p[i] = make_float4(0.f, 0.f, 0.f, 0.f);
}

// ---------------------------------------------------------------------------
// Phase 1: edge scatter.  16 lanes per edge; each lane gathers one float4
// (global_load_b128) of the 64-float message and issues 4 f32 atomic adds.
// x (25.6MB) and s (25.6MB) both fit in the 192MB L2, so gathers hit L2 and
// global_atomic_add_f32 resolves at the L2 atomic units without HBM
// round-trips.  Atomic count (E*64 = 102.4M dwords) is irreducible for the
// unsorted-edge formulation; this kernel minimizes everything around it.
// ---------------------------------------------------------------------------
__global__ void __launch_bounds__(256)
sage_scatter_kernel(const float* __restrict__ x,
                    const long long* __restrict__ ei,   // [2, E] int64
                    float* __restrict__ s,
                    float* __restrict__ cnt) {
    long long gtid = (long long)blockIdx.x * blockDim.x + threadIdx.x;
    int e = (int)(gtid >> 4);
    int q = (int)(gtid & 15);          // quad index: floats [4q, 4q+3]
    if (e >= N_EDGES) return;

    long long srcn = ei[e];            // row 0: message sources
    long long dstn = ei[N_EDGES + e];  // row 1: message targets

    const float4 v = *reinterpret_cast<const float4*>(x + srcn * D + q * 4);
    float* sr = s + dstn * D + q * 4;

    atomicAdd(sr + 0, v.x);
    atomicAdd(sr + 1, v.y);
    atomicAdd(sr + 2, v.z);
    atomicAdd(sr + 3, v.w);
    if (q == 0) atomicAdd(&cnt[dstn], 1.0f);
}

// ---------------------------------------------------------------------------
// Phase 2: fused  out = relu( (s/max(cnt,1)) @ W_l + x @ W_r + b )
// via V_WMMA_F32_16X16X4_F32 (full f32; GEMM is only 1.64 GFLOP so f32
// matrix math is free in this bandwidth-bound pipeline).
//
// Per block: 256 threads = 8 waves; weights staged to LDS in K-PAIRED layout
//   Wls2[kp*64 + col] = (W[2kp][col], W[2kp+1][col])
// so each B fragment is ONE aligned ds_load_b64 straight into an even VGPR
// pair (no repack movs).  Per wave: 16-node x 64-col tile, 4 v8f
// accumulators (initialized with broadcast bias), 16 K-steps x 4 N-tiles x
// 2 weight matrices = 128 WMMA ops.
//
// Fragment layouts (ISA 7.12.2, 32-bit, wave32):
//   A 16x4 :  lane L -> m = L&15;  VGPR v holds A[m][k0 + 2*(L>>4) + v]
//   B 4x16 :  lane L -> n = L&15;  VGPR v holds B[k0 + 2*(L>>4) + v][n]
//   C 16x16:  lane L -> n = L&15;  VGPR v holds C[v + 8*(L>>4)][n]
// ---------------------------------------------------------------------------
__global__ void __launch_bounds__(256)
sage_gemm_kernel(const float* __restrict__ x,
                 const float* __restrict__ s,
                 const float* __restrict__ cnt,
                 const float* __restrict__ Wl,
                 const float* __restrict__ Wr,
                 const float* __restrict__ bias,
                 float* __restrict__ out) {
    __shared__ float2 Wls[32 * D];   // K-paired lin_l weights (16KB)
    __shared__ float2 Wrs[32 * D];   // K-paired lin_r weights (16KB)

    for (int i = threadIdx.x; i < 32 * D; i += 256) {
        const int kp  = i >> 6;      // K-pair index 0..31
        const int col = i & 63;
        Wls[i] = make_float2(Wl[(2 * kp) * D + col], Wl[(2 * kp + 1) * D + col]);
        Wrs[i] = make_float2(Wr[(2 * kp) * D + col], Wr[(2 * kp + 1) * D + col]);
    }
    __syncthreads();

    const int wave    = threadIdx.x >> 5;                 // 0..7
    const int lane    = threadIdx.x & 31;
    const int rowbase = (blockIdx.x * 8 + wave) * 16;     // 16 nodes per wave
    const int m       = lane & 15;                        // A-row / B-col
    const int kk      = lane >> 4;                        // K-pair select

    const int row    = rowbase + m;
    const int rclamp = row < N_NODES ? row : N_NODES - 1; // keep EXEC all-1s

    const float inv = 1.0f / fmaxf(cnt[rclamp], 1.0f);    // mean divisor

    const float* srow = s + (size_t)rclamp * D;
    const float* xrow = x + (size_t)rclamp * D;

    // Accumulators pre-loaded with bias (each output element is accumulated
    // exactly once, so bias folds into C-init).
    v8f acc[4];
    #pragma unroll
    for (int nt = 0; nt < 4; ++nt) {
        const float b0 = bias[nt * 16 + m];
        #pragma unroll
        for (int v = 0; v < 8; ++v) acc[nt][v] = b0;
    }

    #pragma unroll
    for (int kt = 0; kt < 16; ++kt) {
        const int ka = kt * 4 + 2 * kk;   // even -> 8B-aligned pair loads
        const int kp = 2 * kt + kk;       // K-pair row in LDS weight layout

        float2 sv = *reinterpret_cast<const float2*>(srow + ka);
        float2 xv = *reinterpret_cast<const float2*>(xrow + ka);
        v2f a_l; a_l.x = sv.x * inv; a_l.y = sv.y * inv;
        v2f a_r; a_r.x = xv.x;       a_r.y = xv.y;

        #pragma unroll
        for (int nt = 0; nt < 4; ++nt) {
            const int col = nt * 16 + m;
            const v2f b_l = *reinterpret_cast<const v2f*>(&Wls[kp * D + col]);
            const v2f b_r = *reinterpret_cast<const v2f*>(&Wrs[kp * D + col]);

            acc[nt] = __builtin_amdgcn_wmma_f32_16x16x4_f32(
                false, a_l, false, b_l, (short)0, acc[nt], false, false);
            acc[nt] = __builtin_amdgcn_wmma_f32_16x16x4_f32(
                false, a_r, false, b_r, (short)0, acc[nt], false, false);
        }
    }

    // ReLU + store (C/D layout above). Predicate only the stores.
    #pragma unroll
    for (int nt = 0; nt < 4; ++nt) {
        const int col = nt * 16 + m;
        #pragma unroll
        for (int v = 0; v < 8; ++v) {
            const int orow = rowbase + v + 8 * kk;
            if (orow < N_NODES) {
                out[(size_t)orow * D + col] = fmaxf(acc[nt][v], 0.0f);
            }
        }
    }
}

// ---------------------------------------------------------------------------
extern "C" void kernel_launch(void* const* d_in, const int* in_sizes, int n_in,
                              void* d_out, int out_size, void* d_ws, size_t ws_size,
                              hipStream_t stream) {
    const float*     x    = (const float*)d_in[0];
    const long long* ei   = (const long long*)d_in[1];   // int64 [2, E]
    const float*     Wl   = (const float*)d_in[2];
    const float*     Wr   = (const float*)d_in[3];
    const float*     bias = (const float*)d_in[4];
    float*           out  = (float*)d_out;

    // Workspace layout: s[N*64] then cnt[N]  (26 MB).
    float* s   = (float*)d_ws;
    float* cnt = s + (size_t)N_NODES * D;

    const int n4 = (N_NODES * (D + 1)) / 4;
    sage_zero_kernel<<<(n4 + 255) / 256, 256, 0, stream>>>(
        reinterpret_cast<float4*>(s), n4);

    const long long scatter_threads = (long long)N_EDGES * 16;
    sage_scatter_kernel<<<(int)((scatter_threads + 255) / 256), 256, 0, stream>>>(
        x, ei, s, cnt);

    sage_gemm_kernel<<<(N_NODES + 127) / 128, 256, 0, stream>>>(
        x, s, cnt, Wl, Wr, bias, out);
}